// WaveletTransform_23244363006050
// MI455X (gfx1250) — compile-verified
//
#include <hip/hip_runtime.h>
#include <cstdint>

namespace {

constexpr int kN    = 16384;     // input columns per row
constexpr int kM    = 8195;      // outputs per channel = (N + 8 - 1) / 2
constexpr int kOutW = 2 * kM;    // 16390
constexpr int kBlk  = 256;       // 8 wave32 per workgroup

// db4 analysis low-pass taps
__device__ __forceinline__ float lo(int j) {
  constexpr float L[8] = {
    -0.010597401784997278f,  0.032883011666982945f,
     0.030841381835986965f, -0.18703481171888114f,
    -0.02798376941698385f,   0.6308807679295904f,
     0.7148465705525415f,    0.23037781330885523f };
  return L[j];
}

// symmetric extension: x_ext[e] for e in [0, N+13], served from staged LDS row
__device__ __forceinline__ float ext_at(const float* l, int e) {
  int idx = (e < 7) ? (6 - e) : ((e <= kN + 6) ? (e - 7) : (2 * kN + 6 - e));
  return l[idx];
}

// boundary output pair via full symmetric-extension indexing (6 pairs per row total)
__device__ __forceinline__ void boundary_pair(const float* l, int m, float* orow) {
  float a = 0.0f, d = 0.0f;
#pragma unroll
  for (int j = 0; j < 8; ++j) {
    float v = ext_at(l, 2 * m + 8 - j);
    a = fmaf(lo(j), v, a);
    float h = (j & 1) ? lo(7 - j) : -lo(7 - j);  // DEC_HI[j] = (-1)^(j+1) * LO[7-j]
    d = fmaf(h, v, d);
  }
  orow[m]      = a;
  orow[kM + m] = d;
}

// pointer types matching the async-to-LDS builtin's signature:
// (v4i addrspace(1)*, v4i addrspace(3)*, imm offset, imm cpol)
typedef int v4i __attribute__((vector_size(16)));
typedef __attribute__((address_space(1))) v4i* gv4ip;
typedef __attribute__((address_space(3))) v4i* lv4ip;

__global__ __launch_bounds__(kBlk) void dwt_db4_kernel(const float* __restrict__ x,
                                                       float* __restrict__ out) {
  __shared__ float lrow[kN];  // 64 KiB of the 320 KiB WGP LDS

  const int    t    = threadIdx.x;
  const size_t r    = blockIdx.x;
  const float* grow = x + r * (size_t)kN;

  // ---- Phase 1: async DMA the whole row into LDS (CDNA5 async-to-LDS path) ----
  {
#if __has_builtin(__builtin_amdgcn_global_load_async_to_lds_b128)
#pragma unroll
    for (int i = 0; i < kN / (4 * kBlk); ++i) {   // 16 iterations of b128 per lane
      int f4 = t + i * kBlk;                      // float4 index, coalesced across lanes
      __builtin_amdgcn_global_load_async_to_lds_b128(
          (gv4ip)(grow + 4 * f4), (lv4ip)&lrow[4 * f4], 0, 0);
    }
#else
    typedef __attribute__((address_space(3))) float lfloat;
#pragma unroll
    for (int i = 0; i < kN / (4 * kBlk); ++i) {
      int f4 = t + i * kBlk;
      unsigned ldsoff = (unsigned)(unsigned long long)(lfloat*)&lrow[4 * f4];
      const float* ga = grow + 4 * f4;
      asm volatile("global_load_async_to_lds_b128 %0, %1, off"
                   :: "v"(ldsoff), "v"(ga) : "memory");
    }
#endif
#if __has_builtin(__builtin_amdgcn_s_wait_asynccnt)
    __builtin_amdgcn_s_wait_asynccnt(0);
#else
    asm volatile("s_wait_asynccnt 0" ::: "memory");
#endif
  }
  __syncthreads();

  // ---- Phase 2: compute; m = t + 256*i keeps every store wave-coalesced ----
  float* orow = out + r * (size_t)kOutW;

#pragma unroll 4
  for (int i = 0; i < 32; ++i) {
    int m = t + (i << 8);          // covers m in [0, 8191]
    if (m < 3) continue;           // left-mirror pairs handled below (t<3, i==0 only)

    int b = 2 * m - 6;             // even -> 8B-aligned ds_load_b64 reads
    float2 p0 = *(const float2*)&lrow[b];
    float2 p1 = *(const float2*)&lrow[b + 2];
    float2 p2 = *(const float2*)&lrow[b + 4];
    float2 p3 = *(const float2*)&lrow[b + 6];
    float f0 = p0.x, f1 = p0.y, f2 = p1.x, f3 = p1.y;
    float f4 = p2.x, f5 = p2.y, f6 = p3.x, f7 = p3.y;

    // cA[m] = sum_k LO[7-k] * f[k]
    float a =            lo(7) * f0;
    a = fmaf( lo(6), f1, a);  a = fmaf( lo(5), f2, a);  a = fmaf( lo(4), f3, a);
    a = fmaf( lo(3), f4, a);  a = fmaf( lo(2), f5, a);  a = fmaf( lo(1), f6, a);
    a = fmaf( lo(0), f7, a);
    // cD[m] = sum_k (-1)^k LO[k] * f[k]
    float d =            lo(0) * f0;
    d = fmaf(-lo(1), f1, d);  d = fmaf( lo(2), f2, d);  d = fmaf(-lo(3), f3, d);
    d = fmaf( lo(4), f4, d);  d = fmaf(-lo(5), f5, d);  d = fmaf( lo(6), f6, d);
    d = fmaf(-lo(7), f7, d);

    // 256 MB output stream > 192 MB L2 -> nontemporal
    __builtin_nontemporal_store(a, &orow[m]);
    __builtin_nontemporal_store(d, &orow[kM + m]);
  }

  // 6 boundary pairs per row (symmetric extension)
  if (t < 3) {
    boundary_pair(lrow, t, orow);                 // m = 0,1,2
  } else if (t < 6) {
    boundary_pair(lrow, 8192 + (t - 3), orow);    // m = 8192,8193,8194
  }
}

}  // namespace

extern "C" void kernel_launch(void* const* d_in, const int* in_sizes, int n_in,
                              void* d_out, int out_size, void* d_ws, size_t ws_size,
                              hipStream_t stream) {
  (void)n_in; (void)out_size; (void)d_ws; (void)ws_size;
  const float* x   = (const float*)d_in[0];
  float*       out = (float*)d_out;
  int rows = in_sizes[0] / kN;   // 4096 for the reference shapes
  dwt_db4_kernel<<<dim3(rows), dim3(kBlk), 0, stream>>>(x, out);
}